// MultiHeadDynamicSequenceChunker_30468497998168
// MI455X (gfx1250) — compile-verified
//
#include <hip/hip_runtime.h>
#include <hip/hip_bf16.h>
#include <math.h>

#define BATCH 4
#define SEQ   8192
#define DIM   512
#define DQK   512
#define NTOT  1024            // 2*DQK
#define MTOT  (BATCH*SEQ)     // 32768 rows
#define KTOT  DIM             // 512
#define SEG   128             // scan segments per batch
#define TCH   (SEQ/SEG)       // 64 chunks per segment
#define EPSV  1e-8f
#define MBLK  64              // rows per GEMM block
#define NBLKS (MTOT/MBLK)     // 512 blocks
#define BPS   (SEQ/MBLK)      // 128 blocks per sequence

typedef float v2f __attribute__((ext_vector_type(2)));
typedef float v8f __attribute__((ext_vector_type(8)));

// ---------------------------------------------------------------------------
// Kernel 1: fused  qk-GEMM + cosine/probs.  qk never hits HBM.
// Block: 64 rows x all 1024 cols; per n-chunk computes query tile (cols nc*64)
// AND key tile (cols 512+nc*64) together with f32 WMMA 16x16x4; tiles land in
// LDS, per-row norms/dot accumulate via LDS float atomics.  Row0-vs-prev-block
// dot handled by edge export + fixup kernel.
// ---------------------------------------------------------------------------
#define LDA 36
#define LDB 36
#define LDC 65

__global__ __launch_bounds__(256)
void gemm_probs_kernel(const float* __restrict__ tokens,
                       const float* __restrict__ Wqk,
                       float* __restrict__ probs,
                       int* __restrict__ bnd,
                       float* __restrict__ qEdge,
                       float* __restrict__ kEdge,
                       float* __restrict__ qqEdge,
                       float* __restrict__ kkEdge) {
  __shared__ float As[MBLK * LDA];     //  64x32 A tile
  __shared__ float Bs[128 * LDB];      // 128x32 B tile (64 q-rows + 64 k-rows of Wqk)
  __shared__ float Qt[MBLK * LDC];     // 64x64 query result tile
  __shared__ float Kt[MBLK * LDC];     // 64x64 key   result tile
  __shared__ float qqA[MBLK], kkA[MBLK], dotA[MBLK];

  const int tid  = threadIdx.x;
  const int lane = tid & 31;
  const int wave = tid >> 5;
  const int wm   = wave & 3;    // wave M index 0..3
  const int wn   = wave >> 2;   // wave N index 0..1
  const int lm   = lane & 15;
  const int half = lane >> 4;
  const int blk  = blockIdx.x;
  const size_t rowBase = (size_t)blk * MBLK;

  if (tid < MBLK) { qqA[tid] = 0.f; kkA[tid] = 0.f; dotA[tid] = 0.f; }

  const int arow = tid >> 2;          // A loader: 0..63
  const int acol = (tid & 3) * 8;     // 0,8,16,24
  const int brow = tid >> 1;          // B loader: 0..127
  const int bcol = (tid & 1) * 16;    // 0,16

  for (int nc = 0; nc < 8; ++nc) {
    const int colQ = nc * 64;
    const int colK = DQK + nc * 64;
    v8f q0 = {}, q1 = {}, k0 = {}, k1 = {};

    for (int kk = 0; kk < KTOT; kk += 32) {
      const float4 ga0 = *(const float4*)(tokens + (rowBase + arow) * KTOT + kk + acol);
      const float4 ga1 = *(const float4*)(tokens + (rowBase + arow) * KTOT + kk + acol + 4);
      const int wr = (brow < 64) ? (colQ + brow) : (colK + brow - 64);
      const float4 gb0 = *(const float4*)(Wqk + (size_t)wr * KTOT + kk + bcol);
      const float4 gb1 = *(const float4*)(Wqk + (size_t)wr * KTOT + kk + bcol + 4);
      const float4 gb2 = *(const float4*)(Wqk + (size_t)wr * KTOT + kk + bcol + 8);
      const float4 gb3 = *(const float4*)(Wqk + (size_t)wr * KTOT + kk + bcol + 12);
      __syncthreads();   // protect LDS from previous iteration's readers
      *(float4*)(As + arow * LDA + acol)      = ga0;
      *(float4*)(As + arow * LDA + acol + 4)  = ga1;
      *(float4*)(Bs + brow * LDB + bcol)      = gb0;
      *(float4*)(Bs + brow * LDB + bcol + 4)  = gb1;
      *(float4*)(Bs + brow * LDB + bcol + 8)  = gb2;
      *(float4*)(Bs + brow * LDB + bcol + 12) = gb3;
      __syncthreads();

#pragma unroll
      for (int ks = 0; ks < 32; ks += 4) {
        const int ka = ks + 2 * half;
        v2f a;
        a[0] = As[(wm * 16 + lm) * LDA + ka];
        a[1] = As[(wm * 16 + lm) * LDA + ka + 1];
        v2f bq0, bq1, bk0, bk1;
        bq0[0] = Bs[(wn * 32 + lm) * LDB + ka];
        bq0[1] = Bs[(wn * 32 + lm) * LDB + ka + 1];
        bq1[0] = Bs[(wn * 32 + 16 + lm) * LDB + ka];
        bq1[1] = Bs[(wn * 32 + 16 + lm) * LDB + ka + 1];
        bk0[0] = Bs[(64 + wn * 32 + lm) * LDB + ka];
        bk0[1] = Bs[(64 + wn * 32 + lm) * LDB + ka + 1];
        bk1[0] = Bs[(64 + wn * 32 + 16 + lm) * LDB + ka];
        bk1[1] = Bs[(64 + wn * 32 + 16 + lm) * LDB + ka + 1];
        q0 = __builtin_amdgcn_wmma_f32_16x16x4_f32(false, a, false, bq0, (short)0, q0, false, false);
        q1 = __builtin_amdgcn_wmma_f32_16x16x4_f32(false, a, false, bq1, (short)0, q1, false, false);
        k0 = __builtin_amdgcn_wmma_f32_16x16x4_f32(false, a, false, bk0, (short)0, k0, false, false);
        k1 = __builtin_amdgcn_wmma_f32_16x16x4_f32(false, a, false, bk1, (short)0, k1, false, false);
      }
    }

    // dump result tiles to LDS (C/D layout: VGPR v -> row v / v+8)
#pragma unroll
    for (int v = 0; v < 8; ++v) {
      const int r = wm * 16 + half * 8 + v;
      Qt[r * LDC + wn * 32 + lm]      = q0[v];
      Qt[r * LDC + wn * 32 + lm + 16] = q1[v];
      Kt[r * LDC + wn * 32 + lm]      = k0[v];
      Kt[r * LDC + wn * 32 + lm + 16] = k1[v];
    }
    __syncthreads();

    // per-row reductions: 4 threads/row, 16 cols each
    {
      const int r  = tid >> 2;
      const int c0 = (tid & 3) * 16;
      float sqq = 0.f, skk = 0.f, sdot = 0.f;
#pragma unroll
      for (int c = 0; c < 16; ++c) {
        const float qv = Qt[r * LDC + c0 + c];
        const float kv = Kt[r * LDC + c0 + c];
        sqq += qv * qv;
        skk += kv * kv;
        if (r > 0) sdot += qv * Kt[(r - 1) * LDC + c0 + c];
      }
      atomicAdd(&qqA[r], sqq);
      atomicAdd(&kkA[r], skk);
      if (r > 0) atomicAdd(&dotA[r], sdot);
    }
    // edge exports: q row 0 and k row 63 of this block
    if (tid < 64) {
      qEdge[(size_t)blk * DQK + colQ + tid]     = Qt[0 * LDC + tid];
      kEdge[(size_t)blk * DQK + nc * 64 + tid]  = Kt[63 * LDC + tid];
    }
    __syncthreads();
  }

  // finish probs for rows 1..63 (k[n-1] fully in-block)
  if (tid >= 1 && tid < MBLK) {
    const float na  = fmaxf(sqrtf(qqA[tid]), EPSV);
    const float nb  = fmaxf(sqrtf(kkA[tid - 1]), EPSV);
    const float p   = (1.f - dotA[tid] / (na * nb)) * 0.5f;
    const size_t n  = rowBase + tid;
    probs[n] = p;
    bnd[n]   = (p > 0.5f) ? 1 : 0;
  }
  if (tid == 0) { qqEdge[blk] = qqA[0]; kkEdge[blk] = kkA[63]; }
}

// ---------------------------------------------------------------------------
// Kernel 2: fix-up for block-edge rows (one wave per block): row m0 vs k-row
// m0-1 of previous block (or start_key at sequence starts)
// ---------------------------------------------------------------------------
__global__ __launch_bounds__(256)
void edge_probs_kernel(const float* __restrict__ qEdge,
                       const float* __restrict__ kEdge,
                       const float* __restrict__ qqEdge,
                       const float* __restrict__ kkEdge,
                       const float* __restrict__ start_key,
                       float* __restrict__ probs,
                       int* __restrict__ bnd) {
  const int w    = blockIdx.x * (blockDim.x >> 5) + (threadIdx.x >> 5);  // 0..511
  const int lane = threadIdx.x & 31;
  const bool atStart = (w % BPS) == 0;
  const float* q  = qEdge + (size_t)w * DQK;
  const float* kp = atStart ? start_key : (kEdge + (size_t)(w - 1) * DQK);

  float dot = 0.f, kk2 = 0.f;
#pragma unroll
  for (int i = lane * 4; i < DQK; i += 32 * 4) {
    const float4 qv = *(const float4*)(q + i);
    const float4 kv = *(const float4*)(kp + i);
    dot += qv.x * kv.x + qv.y * kv.y + qv.z * kv.z + qv.w * kv.w;
    kk2 += kv.x * kv.x + kv.y * kv.y + kv.z * kv.z + kv.w * kv.w;
  }
#pragma unroll
  for (int m = 16; m > 0; m >>= 1) {
    dot += __shfl_xor(dot, m, 32);
    kk2 += __shfl_xor(kk2, m, 32);
  }
  if (lane == 0) {
    const float na  = fmaxf(sqrtf(qqEdge[w]), EPSV);
    const float nb2 = atStart ? kk2 : kkEdge[w - 1];
    const float nb  = fmaxf(sqrtf(nb2), EPSV);
    const float p   = (1.f - dot / (na * nb)) * 0.5f;
    const size_t n  = (size_t)w * MBLK;
    probs[n] = p;
    bnd[n]   = (p > 0.5f || atStart) ? 1 : 0;
  }
}

// ---------------------------------------------------------------------------
// Kernel 3: per-batch block scan -> rank, compacted pos[] and chunk probs
// ---------------------------------------------------------------------------
__global__ __launch_bounds__(1024)
void rank_kernel(const float* __restrict__ probs,
                 const int* __restrict__ bnd,
                 int* __restrict__ rank,
                 int* __restrict__ pos,
                 float* __restrict__ chunk_p) {
  const int b = blockIdx.x;
  const int t = threadIdx.x;
  const int base = b * SEQ;
  const int per = SEQ / 1024;  // 8

  for (int i = t; i < SEQ; i += 1024) { pos[base + i] = 0; chunk_p[base + i] = 0.f; }
  __threadfence_block();

  int flags[per];
  int lsum = 0;
  const int n0 = t * per;
#pragma unroll
  for (int e = 0; e < per; ++e) { flags[e] = bnd[base + n0 + e]; lsum += flags[e]; }

  __shared__ int ssum[1024];
  ssum[t] = lsum;
  __syncthreads();
  for (int off = 1; off < 1024; off <<= 1) {
    const int add = (t >= off) ? ssum[t - off] : 0;
    __syncthreads();
    ssum[t] += add;
    __syncthreads();
  }
  int run = ssum[t] - lsum;  // exclusive prefix

#pragma unroll
  for (int e = 0; e < per; ++e) {
    const int n = n0 + e;
    run += flags[e];
    rank[base + n] = run - 1;
    if (flags[e]) {
      pos[base + run - 1] = n;
      chunk_p[base + run - 1] = probs[base + n];
    }
  }
}

// ---------------------------------------------------------------------------
// Kernel 4: per-segment local scan, fused with writing downsampled.
// ---------------------------------------------------------------------------
__global__ __launch_bounds__(256)
void segscan_kernel(const float* __restrict__ tokens,
                    const int* __restrict__ pos,
                    const float* __restrict__ chunk_p,
                    float* __restrict__ out_ds,
                    float* __restrict__ local_scan,
                    float* __restrict__ pg_arr,
                    float* __restrict__ A_seg,
                    float* __restrict__ B_seg) {
  const int b = blockIdx.y;
  const int s = blockIdx.x;
  const int t = threadIdx.x;
  const int d = t * 2;
  const size_t base = (size_t)b * SEQ;

  float2 h = make_float2(0.f, 0.f);
  float pg = 1.f;
  int j = s * TCH;
  float p = chunk_p[base + j];
  int row = pos[base + j];

  for (int jj = 0; jj < TCH; ++jj) {
    float pN = 0.f; int rowN = 0;
    if (jj + 1 < TCH) {
      pN = chunk_p[base + j + 1];
      rowN = pos[base + j + 1];
      __builtin_prefetch((const void*)(tokens + (base + rowN) * DIM + d), 0, 0);
    }
    const float2 tok = *(const float2*)(tokens + (base + row) * DIM + d);
    const float g = 1.f - p;
    const float2 ds = make_float2(tok.x * p, tok.y * p);
    h.x = g * h.x + ds.x;
    h.y = g * h.y + ds.y;
    *(float2*)(out_ds     + (base + j) * DIM + d) = ds;
    *(float2*)(local_scan + (base + j) * DIM + d) = h;
    pg *= g;
    if (t == 0) pg_arr[base + j] = pg;
    p = pN; row = rowN; ++j;
  }
  *(float2*)(B_seg + ((size_t)(b * SEG + s)) * DIM + d) = h;
  if (t == 0) A_seg[b * SEG + s] = pg;
}

// ---------------------------------------------------------------------------
// Kernel 5: serial carry scan over segment summaries (SEG=128 steps)
// ---------------------------------------------------------------------------
__global__ __launch_bounds__(256)
void carry_kernel(const float* __restrict__ A_seg,
                  const float* __restrict__ B_seg,
                  float* __restrict__ carryIn) {
  const int b = blockIdx.x;
  const int t = threadIdx.x;
  const int d = t * 2;
  float2 c = make_float2(0.f, 0.f);
  for (int s = 0; s < SEG; ++s) {
    const size_t idx = ((size_t)(b * SEG + s)) * DIM + d;
    *(float2*)(carryIn + idx) = c;
    const float a = A_seg[b * SEG + s];
    const float2 Bv = *(const float2*)(B_seg + idx);
    c.x = a * c.x + Bv.x;
    c.y = a * c.y + Bv.y;
  }
}

// ---------------------------------------------------------------------------
// Kernel 6: upsample gather fused with carry fix-up
// ---------------------------------------------------------------------------
__global__ __launch_bounds__(256)
void upsample_kernel(const float* __restrict__ local_scan,
                     const float* __restrict__ pg_arr,
                     const float* __restrict__ carryIn,
                     const int* __restrict__ rank,
                     float* __restrict__ out_up) {
  const int b = blockIdx.y;
  const int n = blockIdx.x;
  const int t = threadIdx.x;
  const int d = t * 2;
  const size_t base = (size_t)b * SEQ;
  const int r = rank[base + n];
  const int s = r / TCH;
  const float pg = pg_arr[base + r];
  const float2 loc = *(const float2*)(local_scan + (base + r) * DIM + d);
  const float2 c   = *(const float2*)(carryIn + ((size_t)(b * SEG + s)) * DIM + d);
  *(float2*)(out_up + (base + n) * DIM + d) =
      make_float2(loc.x + pg * c.x, loc.y + pg * c.y);
}

// ---------------------------------------------------------------------------
// Kernel 7: aux loss scalar
// ---------------------------------------------------------------------------
__global__ __launch_bounds__(256)
void aux_kernel(const float* __restrict__ probs,
                const int* __restrict__ bnd,
                float* __restrict__ out_scalar) {
  __shared__ float sG[256], sF[256];
  const int t = threadIdx.x;
  float aux_acc = 0.f;
  for (int b = 0; b < BATCH; ++b) {
    float gp = 0.f, fp = 0.f;
    for (int n = t; n < SEQ; n += 256) {
      gp += probs[b * SEQ + n];
      fp += (float)bnd[b * SEQ + n];
    }
    sG[t] = gp; sF[t] = fp;
    __syncthreads();
    for (int off = 128; off > 0; off >>= 1) {
      if (t < off) { sG[t] += sG[t + off]; sF[t] += sF[t + off]; }
      __syncthreads();
    }
    if (t == 0) {
      const float Gm = sG[0] / SEQ;
      const float Fm = sF[0] / SEQ;
      const float Nt = 6.f;
      aux_acc += Nt / (Nt - 1.f) * ((Nt - 1.f) * Fm * Gm + (1.f - Fm) * (1.f - Gm));
    }
    __syncthreads();
  }
  if (t == 0) out_scalar[0] = aux_acc / BATCH * 0.03f;
}

// ---------------------------------------------------------------------------
extern "C" void kernel_launch(void* const* d_in, const int* in_sizes, int n_in,
                              void* d_out, int out_size, void* d_ws, size_t ws_size,
                              hipStream_t stream) {
  const float* tokens    = (const float*)d_in[0];
  const float* Wqk       = (const float*)d_in[1];
  const float* start_key = (const float*)d_in[2];

  float* out        = (float*)d_out;
  float* out_ds     = out;
  float* out_up     = out + (size_t)MTOT * DIM;
  float* out_scalar = out + (size_t)2 * MTOT * DIM;

  char* w = (char*)d_ws;
  float* local_scan = (float*)w; w += (size_t)MTOT * DIM * sizeof(float);       // 64 MB
  float* qEdge      = (float*)w; w += (size_t)NBLKS * DQK * sizeof(float);      //  1 MB
  float* kEdge      = (float*)w; w += (size_t)NBLKS * DQK * sizeof(float);      //  1 MB
  float* B_seg      = (float*)w; w += (size_t)BATCH * SEG * DIM * sizeof(float);
  float* carryIn    = (float*)w; w += (size_t)BATCH * SEG * DIM * sizeof(float);
  float* probs      = (float*)w; w += (size_t)MTOT * sizeof(float);
  float* chunk_p    = (float*)w; w += (size_t)MTOT * sizeof(float);
  float* pg_arr     = (float*)w; w += (size_t)MTOT * sizeof(float);
  float* A_seg      = (float*)w; w += (size_t)BATCH * SEG * sizeof(float);
  float* qqEdge     = (float*)w; w += (size_t)NBLKS * sizeof(float);
  float* kkEdge     = (float*)w; w += (size_t)NBLKS * sizeof(float);
  int*   bnd        = (int*)w;   w += (size_t)MTOT * sizeof(int);
  int*   rank       = (int*)w;   w += (size_t)MTOT * sizeof(int);
  int*   pos        = (int*)w;   w += (size_t)MTOT * sizeof(int);

  gemm_probs_kernel<<<dim3(NBLKS), 256, 0, stream>>>(tokens, Wqk, probs, bnd,
                                                     qEdge, kEdge, qqEdge, kkEdge);
  edge_probs_kernel<<<dim3(NBLKS / 8), 256, 0, stream>>>(qEdge, kEdge, qqEdge, kkEdge,
                                                         start_key, probs, bnd);
  rank_kernel<<<dim3(BATCH), 1024, 0, stream>>>(probs, bnd, rank, pos, chunk_p);
  segscan_kernel<<<dim3(SEG, BATCH), 256, 0, stream>>>(tokens, pos, chunk_p,
                                                       out_ds, local_scan, pg_arr,
                                                       A_seg, B_seg);
  carry_kernel<<<dim3(BATCH), 256, 0, stream>>>(A_seg, B_seg, carryIn);
  upsample_kernel<<<dim3(SEQ, BATCH), 256, 0, stream>>>(local_scan, pg_arr, carryIn,
                                                        rank, out_up);
  aux_kernel<<<1, 256, 0, stream>>>(probs, bnd, out_scalar);
}